// GCN_25563645346635
// MI455X (gfx1250) — compile-verified
//
#include <hip/hip_runtime.h>
#include <math.h>

typedef float v2f __attribute__((ext_vector_type(2)));
typedef float v8f __attribute__((ext_vector_type(8)));

#define HIDDEN  128
#define NROIS   400
#define NGRAPHS 512

#define AS1 __attribute__((address_space(1)))
#define AS3 __attribute__((address_space(3)))

// exact pointee type expected by the async-LDS builtin (per hipcc diagnostic:
// "__attribute__((__vector_size__(4 * sizeof(int)))) int __device__ *")
typedef int v4i_vs __attribute__((vector_size(16)));

#if __has_builtin(__builtin_amdgcn_global_load_async_to_lds_b128) && \
    __has_builtin(__builtin_amdgcn_s_wait_asynccnt)
#define USE_ASYNC_LDS 1
#endif

// ---------- init: deg = 1.0 (self loop), pooled = 0 (ReLU lower bound) ----------
__global__ void k_init(float* __restrict__ deg, float* __restrict__ pooled, int N) {
    int i = blockIdx.x * blockDim.x + threadIdx.x;
    if (i < N) deg[i] = 1.0f;
    if (i < NGRAPHS * HIDDEN) pooled[i] = 0.0f;
}

// ---------- in-degree accumulation over targets (col) ----------
__global__ void k_deg(const int* __restrict__ col, float* __restrict__ deg, int E) {
    int e = blockIdx.x * blockDim.x + threadIdx.x;
    if (e < E) atomicAdd(&deg[col[e]], 1.0f);
}

// ---------- dinv = rsqrt(deg), in place ----------
__global__ void k_dinv(float* __restrict__ deg, int N) {
    int i = blockIdx.x * blockDim.x + threadIdx.x;
    if (i < N) {
        float d = deg[i];
        deg[i] = (d > 0.0f) ? rsqrtf(d) : 0.0f;
    }
}

// ---------- h = x @ W via V_WMMA_F32_16X16X4_F32 ----------
// Block = 256 threads = 8 waves. Block b owns rows [16b, 16b+16); wave w owns
// output columns [16w, 16w+16). x tile (16x400) staged once in LDS (async
// global->LDS DMA when available); W served from L2 (200 KB, shared by all
// 6250 blocks).
__global__ void __launch_bounds__(256)
k_gemm(const float* __restrict__ x, const float* __restrict__ W,
       float* __restrict__ h, int N) {
    __shared__ __align__(16) float xs[16][NROIS + 4];   // +4 dwords: kill bank conflicts
    const int  tid  = threadIdx.x;
    const long row0 = (long)blockIdx.x * 16;

    // stage 16x400 x-tile (1600 x 16B / 256 threads)
    for (int i = tid; i < 16 * (NROIS / 4); i += 256) {
        int r  = i / (NROIS / 4);
        int c4 = i % (NROIS / 4);
        const float* gp = x + (row0 + r) * (long)NROIS + c4 * 4;
        float*       lp = &xs[r][c4 * 4];
        if (row0 + r < N) {
#ifdef USE_ASYNC_LDS
            __builtin_amdgcn_global_load_async_to_lds_b128(
                (AS1 v4i_vs*)gp, (AS3 v4i_vs*)lp, /*offset=*/0, /*cpol=*/0);
#else
            *(float4*)lp = *(const float4*)gp;
#endif
        } else {
            *(float4*)lp = make_float4(0.f, 0.f, 0.f, 0.f);
        }
    }
#ifdef USE_ASYNC_LDS
    __builtin_amdgcn_s_wait_asynccnt(0);
#endif
    __syncthreads();

    const int wave = tid >> 5;
    const int lane = tid & 31;
    const int n0   = wave * 16;
    const int lr   = lane & 15;   // M for A / N for B,C,D
    const int hi   = lane >> 4;   // K-pair selector for A/B, M-half for C/D

    v8f acc = {0.f, 0.f, 0.f, 0.f, 0.f, 0.f, 0.f, 0.f};

#pragma unroll 4
    for (int k = 0; k < NROIS; k += 4) {
        const int ka = k + hi * 2;
        v2f a;                                   // A 16x4: lanes0-15 K={0,1}, lanes16-31 K={2,3}
        a.x = xs[lr][ka];
        a.y = xs[lr][ka + 1];
        v2f bf;                                  // B 4x16: mirrored layout, N = lr
        bf.x = W[(ka + 0) * HIDDEN + n0 + lr];
        bf.y = W[(ka + 1) * HIDDEN + n0 + lr];
        acc = __builtin_amdgcn_wmma_f32_16x16x4_f32(
            /*neg_a=*/false, a, /*neg_b=*/false, bf,
            /*c_mod=*/(short)0, acc, /*reuse_a=*/false, /*reuse_b=*/false);
    }

    // C/D layout: VGPR j -> M = 8*hi + j, N = lr
    const int nCol = n0 + lr;
    if (row0 + 16 <= (long)N) {
        // full tile: uniform branch, 32-bit addressing, no per-element exec dance
        float* hp = h + (int)(row0 + hi * 8) * HIDDEN + nCol;
#pragma unroll
        for (int j = 0; j < 8; ++j) hp[j * HIDDEN] = acc[j];
    } else {
#pragma unroll
        for (int j = 0; j < 8; ++j) {
            long m = row0 + hi * 8 + j;
            if (m < N) h[m * HIDDEN + nCol] = acc[j];
        }
    }
}

// ---------- self-loop term: agg[n] = h[n] * dinv[n]^2 ----------
__global__ void k_self(const float* __restrict__ h, const float* __restrict__ dinv,
                       float* __restrict__ agg, int N) {
    long idx  = (long)blockIdx.x * blockDim.x + threadIdx.x;
    long node = idx >> 5;
    int  lane = (int)(idx & 31);
    if (node >= N) return;
    float w  = dinv[node]; w *= w;
    float4 v = ((const float4*)(h + node * HIDDEN))[lane];
    float4 o = make_float4(v.x * w, v.y * w, v.z * w, v.w * w);
    ((float4*)(agg + node * HIDDEN))[lane] = o;
}

// ---------- edge scatter: one wave per edge, 4 hidden dims per lane ----------
__global__ void k_edge(const int* __restrict__ row, const int* __restrict__ col,
                       const float* __restrict__ dinv, const float* __restrict__ h,
                       float* __restrict__ agg, int E) {
    long gtid = (long)blockIdx.x * blockDim.x + threadIdx.x;
    long e    = gtid >> 5;
    int  lane = (int)(gtid & 31);
    if (e >= E) return;
    int r = row[e], c = col[e];
    float w  = dinv[r] * dinv[c];
    float4 v = ((const float4*)(h + (long)r * HIDDEN))[lane];
    float* dst = agg + (long)c * HIDDEN + lane * 4;
    atomicAdd(dst + 0, v.x * w);
    atomicAdd(dst + 1, v.y * w);
    atomicAdd(dst + 2, v.z * w);
    atomicAdd(dst + 3, v.w * w);
}

// ---------- fused bias + ReLU + segment-max pool (u32 atomicMax, vals >= 0) ----------
__global__ void k_pool(const float* __restrict__ agg, const float* __restrict__ bias,
                       const int* __restrict__ batch, float* __restrict__ pooled, int N) {
    long gtid = (long)blockIdx.x * blockDim.x + threadIdx.x;
    long node = gtid >> 5;
    int  lane = (int)(gtid & 31);
    if (node >= N) return;
    int g = batch[node];
    float4 v  = ((const float4*)(agg + node * HIDDEN))[lane];
    float4 bb = ((const float4*)bias)[lane];
    unsigned* dst = (unsigned*)(pooled + (long)g * HIDDEN) + lane * 4;
    atomicMax(dst + 0, __float_as_uint(fmaxf(v.x + bb.x, 0.0f)));
    atomicMax(dst + 1, __float_as_uint(fmaxf(v.y + bb.y, 0.0f)));
    atomicMax(dst + 2, __float_as_uint(fmaxf(v.z + bb.z, 0.0f)));
    atomicMax(dst + 3, __float_as_uint(fmaxf(v.w + bb.w, 0.0f)));
}

// ---------- logits = pooled @ lin_W + lin_b  (512x128x2, trivial) ----------
__global__ void k_logits(const float* __restrict__ pooled, const float* __restrict__ lw,
                         const float* __restrict__ lb, float* __restrict__ logits) {
    int g = blockIdx.x * blockDim.x + threadIdx.x;
    if (g >= NGRAPHS) return;
    float s0 = lb[0], s1 = lb[1];
#pragma unroll 8
    for (int d = 0; d < HIDDEN; ++d) {
        float p = pooled[(long)g * HIDDEN + d];
        s0 += p * lw[d * 2 + 0];
        s1 += p * lw[d * 2 + 1];
    }
    logits[g * 2 + 0] = s0;
    logits[g * 2 + 1] = s1;
}

extern "C" void kernel_launch(void* const* d_in, const int* in_sizes, int n_in,
                              void* d_out, int out_size, void* d_ws, size_t ws_size,
                              hipStream_t stream) {
    const float* x     = (const float*)d_in[0];
    const int*   ei    = (const int*)d_in[1];
    const int*   batch = (const int*)d_in[2];
    const float* W     = (const float*)d_in[3];
    const float* b     = (const float*)d_in[4];
    const float* lw    = (const float*)d_in[5];
    const float* lb    = (const float*)d_in[6];

    const int N = in_sizes[2];        // 100000
    const int E = in_sizes[1] / 2;    // 1600000
    const int* rowi = ei;             // edge_index[0] = source
    const int* coli = ei + E;         // edge_index[1] = target

    float* logits = (float*)d_out;                    // [512*2]
    float* pooled = (float*)d_out + NGRAPHS * 2;      // [512*128]

    float* h   = (float*)d_ws;                        // N*128 f32
    float* agg = h + (size_t)N * HIDDEN;              // N*128 f32
    float* deg = agg + (size_t)N * HIDDEN;            // N f32 (becomes dinv)

    const int T = 256;
    const int initN = (N > NGRAPHS * HIDDEN) ? N : NGRAPHS * HIDDEN;

    k_init  <<<(initN + T - 1) / T, T, 0, stream>>>(deg, pooled, N);
    k_deg   <<<(E + T - 1) / T, T, 0, stream>>>(coli, deg, E);
    k_dinv  <<<(N + T - 1) / T, T, 0, stream>>>(deg, N);
    k_gemm  <<<(N + 15) / 16, T, 0, stream>>>(x, W, h, N);
    k_self  <<<(int)(((long)N * 32 + T - 1) / T), T, 0, stream>>>(h, deg, agg, N);
    k_edge  <<<(int)(((long)E * 32 + T - 1) / T), T, 0, stream>>>(rowi, coli, deg, h, agg, E);
    k_pool  <<<(int)(((long)N * 32 + T - 1) / T), T, 0, stream>>>(agg, b, batch, pooled, N);
    k_logits<<<(NGRAPHS + T - 1) / T, T, 0, stream>>>(pooled, lw, lb, logits);
}